// DCM_38113539784875
// MI455X (gfx1250) — compile-verified
//
#include <hip/hip_runtime.h>

typedef __attribute__((ext_vector_type(2))) float v2f;
typedef __attribute__((ext_vector_type(4))) float v4f;
typedef __attribute__((ext_vector_type(8))) float v8f;

#define NQ 16384
#define MR 16384
#define DIMK 16
#define NSPLIT 4                 // M-dimension splits per query tile
#define QT_PER_BLK 2             // query tiles per block
#define COLS_PER_SPLIT (MR / NSPLIT)          // 4096
#define ITER_PER_SPLIT (COLS_PER_SPLIT / 32)  // 128 (32 cols/iter)

// ---------- Prologue: xb2[j] = |xb_j|^2 (1 MB read, negligible) ----------
__global__ __launch_bounds__(256) void DCM_xb2_kernel(const float* __restrict__ xb,
                                                      float* __restrict__ xb2) {
  const int j = blockIdx.x * 256 + threadIdx.x;
  const v4f* row = (const v4f*)(xb + (size_t)j * DIMK);
  v4f r0 = row[0], r1 = row[1], r2 = row[2], r3 = row[3];
  float s = r0.x * r0.x + r0.y * r0.y + r0.z * r0.z + r0.w * r0.w;
  s += r1.x * r1.x + r1.y * r1.y + r1.z * r1.z + r1.w * r1.w;
  s += r2.x * r2.x + r2.y * r2.y + r2.z * r2.z + r2.w * r2.w;
  s += r3.x * r3.x + r3.y * r3.y + r3.z * r3.z + r3.w * r3.w;
  xb2[j] = s;
}

// ---------- Main: 8 waves/block = 2 query tiles x 4 M-splits ----------
// Score[i,j] = |xb_j|^2 - 2<x_i,xb_j>  (|x_i|^2 constant per row -> dropped).
// WMMA computes D = (-2*X) * XB^T + C with C pre-seeded to |xb_j|^2.
__global__ __launch_bounds__(256) void DCM_nn_wmma_kernel(
    const float* __restrict__ x,
    const float* __restrict__ xb,
    const float* __restrict__ y,
    const float* __restrict__ xb2,
    float* __restrict__ out) {
  __shared__ float s_val[QT_PER_BLK][NSPLIT][16];
  __shared__ int   s_idx[QT_PER_BLK][NSPLIT][16];

  const int wave  = threadIdx.x >> 5;        // 0..7
  const int lane  = threadIdx.x & 31;
  const int qtloc = wave >> 2;               // 0..1 : query tile within block
  const int split = wave & 3;                // 0..3 : M-range split
  const int qbase = (blockIdx.x * QT_PER_BLK + qtloc) * 16;
  const int grp   = lane & 15;               // row (A) / col (B,C,D) in 16x16 tile
  const int hi    = lane >> 4;               // 0 -> K pair (0,1); 1 -> K pair (2,3)

  // A fragment: 16x4 f32 per K-step, pre-scaled by -2.
  v2f a[4];
  {
    const float* xrow = x + (size_t)(qbase + grp) * DIMK;
#pragma unroll
    for (int k = 0; k < 4; ++k) {
      const int kb = k * 4 + hi * 2;
      a[k].x = -2.0f * xrow[kb + 0];
      a[k].y = -2.0f * xrow[kb + 1];
    }
  }

  float bv0[8], bv1[8];
  int   bi0[8], bi1[8];
#pragma unroll
  for (int r = 0; r < 8; ++r) {
    bv0[r] = bv1[r] = 3.402823466e38f;
    bi0[r] = bi1[r] = 0;
  }

  const int jt0 = split * ITER_PER_SPLIT;
  for (int jt = jt0; jt < jt0 + ITER_PER_SPLIT; ++jt) {
    const int c0 = jt * 32 + grp;            // this lane's column, stream 0
    const int c1 = c0 + 16;                  // stream 1
    const float* p0 = xb + (size_t)c0 * DIMK;
    const float* p1 = xb + (size_t)c1 * DIMK;

    // B fragments (4x16 per K-step): lane holds col, same K split as A.
    v2f b0[4], b1[4];
#pragma unroll
    for (int k = 0; k < 4; ++k) {
      const int kb = k * 4 + hi * 2;
      b0[k] = *(const v2f*)(p0 + kb);
      b1[k] = *(const v2f*)(p1 + kb);
    }
    const float s0 = xb2[c0];
    const float s1 = xb2[c1];

    // Seed C with |xb_j|^2, then two independent 4-deep WMMA chains.
    v8f acc0, acc1;
#pragma unroll
    for (int r = 0; r < 8; ++r) { acc0[r] = s0; acc1[r] = s1; }
#pragma unroll
    for (int k = 0; k < 4; ++k) {
      acc0 = __builtin_amdgcn_wmma_f32_16x16x4_f32(
          false, a[k], false, b0[k], (short)0, acc0, false, false);
      acc1 = __builtin_amdgcn_wmma_f32_16x16x4_f32(
          false, a[k], false, b1[k], (short)0, acc1, false, false);
    }

    // Dual running argmin. acc[r]: row = r + hi*8, col = c0 / c1.
#pragma unroll
    for (int r = 0; r < 8; ++r) {
      if (acc0[r] < bv0[r]) { bv0[r] = acc0[r]; bi0[r] = c0; }
      if (acc1[r] < bv1[r]) { bv1[r] = acc1[r]; bi1[r] = c1; }
    }
  }

  // Merge the two streams (index tie-break keeps first-occurrence argmin).
  float bestv[8];
  int   besti[8];
#pragma unroll
  for (int r = 0; r < 8; ++r) {
    const bool take1 = (bv1[r] < bv0[r]) || (bv1[r] == bv0[r] && bi1[r] < bi0[r]);
    bestv[r] = take1 ? bv1[r] : bv0[r];
    besti[r] = take1 ? bi1[r] : bi0[r];
  }

  // Min+index butterfly across the 16-lane column group (stays in half-wave).
#pragma unroll
  for (int mask = 1; mask < 16; mask <<= 1) {
#pragma unroll
    for (int r = 0; r < 8; ++r) {
      const float ov = __shfl_xor(bestv[r], mask, 32);
      const int   oi = __shfl_xor(besti[r], mask, 32);
      if (ov < bestv[r] || (ov == bestv[r] && oi < besti[r])) {
        bestv[r] = ov; besti[r] = oi;
      }
    }
  }

  // Lane 0 posts rows 0..7, lane 16 posts rows 8..15 for this split.
  if (grp == 0) {
#pragma unroll
    for (int r = 0; r < 8; ++r) {
      s_val[qtloc][split][hi * 8 + r] = bestv[r];
      s_idx[qtloc][split][hi * 8 + r] = besti[r];
    }
  }
  __syncthreads();

  // Fold the 4 splits (ascending split order == ascending index order,
  // so strict < preserves first-index argmin), gather y, store.
  if (threadIdx.x < QT_PER_BLK * 16) {
    const int qt  = threadIdx.x >> 4;
    const int row = threadIdx.x & 15;
    float v = s_val[qt][0][row];
    int   i = s_idx[qt][0][row];
#pragma unroll
    for (int s = 1; s < NSPLIT; ++s) {
      const float sv = s_val[qt][s][row];
      const int   si = s_idx[qt][s][row];
      if (sv < v) { v = sv; i = si; }
    }
    out[(blockIdx.x * QT_PER_BLK + qt) * 16 + row] = y[i];
  }
}

extern "C" void kernel_launch(void* const* d_in, const int* in_sizes, int n_in,
                              void* d_out, int out_size, void* d_ws, size_t ws_size,
                              hipStream_t stream) {
  const float* x  = (const float*)d_in[0];   // [16384, 16] queries
  const float* xb = (const float*)d_in[1];   // [16384, 16] reference set
  const float* y  = (const float*)d_in[2];   // [16384] labels
  float* out = (float*)d_out;                // [16384]
  float* xb2 = (float*)d_ws;                 // 16384 floats = 64 KB scratch

  DCM_xb2_kernel<<<dim3(MR / 256), dim3(256), 0, stream>>>(xb, xb2);

  // 1024 query tiles; 2 tiles x 4 splits (8 waves, 256 threads) per block.
  DCM_nn_wmma_kernel<<<dim3(NQ / 16 / QT_PER_BLK), dim3(256), 0, stream>>>(
      x, xb, y, xb2, out);
}